// ElementaryCA_31602369364360
// MI455X (gfx1250) — compile-verified
//
#include <hip/hip_runtime.h>
#include <stdint.h>
#include <stddef.h>

// Problem constants (from the reference setup)
#define RB 16            // batch rows
#define RW 32768         // cells per row (power of two -> wrap via mask)
#define RSTEPS 128       // CA steps
#define TS (RSTEPS + 1)  // output time slices (includes t=0)

// Blocking
#define BLOCK 128                       // threads per block (4 waves)
#define NWAVES (BLOCK / 32)
#define CPL 16                          // output cells per lane
#define SEG (BLOCK * CPL)               // 2048 cells per block
#define HALO RSTEPS                     // 128-cell halo each side
#define WIN_WORDS ((SEG + 2 * HALO) / 32)  // 72 packed words per block window
#define WPW (WIN_WORDS / NWAVES)        // 18 words packed per wave
#define NWIN 9                          // 288-cell register window per lane
#define NV4 (CPL / 4)                   // float4 stores per lane per slice
#define NBLOCKS (RB * (RW / SEG))       // 256 blocks

// One CA step on a 9-word bit-packed window held in registers.
// Bit b of word i == cell (i*32 + b); left neighbor = bit-1 (LSB-first).
template <bool FAST>
__device__ __forceinline__ void ca_step(uint32_t w[NWIN], const uint32_t mk[8]) {
    uint32_t nw[NWIN];
#pragma unroll
    for (int i = 0; i < NWIN; ++i) {
        const uint32_t wl = (i > 0) ? w[i - 1] : 0u;
        const uint32_t wr = (i < NWIN - 1) ? w[i + 1] : 0u;
        const uint32_t C = w[i];
        const uint32_t L = (C << 1) | (wl >> 31);
        const uint32_t R = (C >> 1) | (wr << 31);
        if (FAST) {
            // Rule 110: new = (C ^ R) | (C & ~L)
            nw[i] = (C ^ R) | (C & ~L);
        } else {
            // Generic 8-entry LUT via bit-sliced mux tree (select bits R,C,L)
            const uint32_t x0 = mk[0] ^ (R & (mk[0] ^ mk[1]));
            const uint32_t x1 = mk[2] ^ (R & (mk[2] ^ mk[3]));
            const uint32_t x2 = mk[4] ^ (R & (mk[4] ^ mk[5]));
            const uint32_t x3 = mk[6] ^ (R & (mk[6] ^ mk[7]));
            const uint32_t y0 = x0 ^ (C & (x0 ^ x1));
            const uint32_t y1 = x2 ^ (C & (x2 ^ x3));
            nw[i] = y0 ^ (L & (y0 ^ y1));
        }
    }
#pragma unroll
    for (int i = 0; i < NWIN; ++i) w[i] = nw[i];
}

template <bool FAST>
__device__ __forceinline__ void time_loop(uint32_t w[NWIN], const uint32_t mk[8],
                                          float4* __restrict__ stage4,  // __shared__, 2*BLOCK*NV4
                                          uint32_t lds_base, float* __restrict__ outb,
                                          int tid, int bitoff) {
    const uint32_t wv = (uint32_t)(tid >> 5);
    const uint32_t ln = (uint32_t)(tid & 31);
#pragma unroll 1
    for (int s = 0; s < TS; ++s) {
        const int buf = s & 1;
        // Ensure this LDS buffer's async stores from slice s-2 have drained
        // (per-wave ASYNCcnt; 4 in flight from slice s-1 may remain).
        asm volatile("s_wait_asynccnt 0x4" ::: "memory");

        // Unpack this lane's 16 cells (center word of the window) to LDS stage.
        const uint32_t cw = w[4] >> bitoff;
        float4* st = stage4 + (size_t)(buf * BLOCK + tid) * NV4;
#pragma unroll
        for (int k = 0; k < NV4; ++k) {
            float4 f;
            f.x = (float)((cw >> (4 * k + 0)) & 1u);
            f.y = (float)((cw >> (4 * k + 1)) & 1u);
            f.z = (float)((cw >> (4 * k + 2)) & 1u);
            f.w = (float)((cw >> (4 * k + 3)) & 1u);
            st[k] = f;
        }
        // Async engine must observe the DS stores (same wave only is read back).
        asm volatile("s_wait_dscnt 0x0" ::: "memory");

        // 4 async LDS->global stores per wave, each 512B contiguous, streamed
        // non-temporally (write-once output, never re-read; 270MB > 192MB L2):
        // instr k covers bytes [(wv*4+k)*512, +512) of this block's slice segment.
        const uint32_t sliceb = (uint32_t)s * (RW * 4u);
#pragma unroll
        for (int k = 0; k < NV4; ++k) {
            const uint32_t voff = sliceb + (wv * 4u + (uint32_t)k) * 512u + ln * 16u;
            const uint32_t lofs = lds_base + (uint32_t)buf * (BLOCK * CPL * 4u) +
                                  wv * 2048u + (uint32_t)k * 512u + ln * 16u;
            asm volatile("global_store_async_from_lds_b128 %0, %1, %2 th:TH_STORE_NT"
                         :: "v"(voff), "v"(lofs), "s"(outb)
                         : "memory");
        }

        if (s < RSTEPS) ca_step<FAST>(w, mk);
    }
    asm volatile("s_wait_asynccnt 0x0" ::: "memory");
}

__global__ __launch_bounds__(BLOCK) void ca110_kernel(const float* __restrict__ state,
                                                      const float* __restrict__ rtab,
                                                      float* __restrict__ out) {
    __shared__ uint32_t words[WIN_WORDS];
    __shared__ float4 stage[2 * BLOCK * NV4];  // 16 KB double-buffered stage

    const int tid = threadIdx.x;
    const int row = blockIdx.x / (RW / SEG);
    const int seg = (blockIdx.x % (RW / SEG)) * SEG;

    // Rule table -> per-pattern bit masks (uniform across lanes; compiler
    // hoists these to scalar loads)
    uint32_t mk[8];
    uint32_t rbyte = 0;
#pragma unroll
    for (int i = 0; i < 8; ++i) {
        const uint32_t b = (rtab[i] != 0.0f) ? 1u : 0u;
        mk[i] = 0u - b;
        rbyte |= b << i;
    }

    // Pack phase, latency-hidden: issue all 18 coalesced loads first (all in
    // flight together), then ballot each 32-cell group into a packed word.
    const float* srow = state + (size_t)row * RW;
    const int lane = tid & 31;
    const int wave = tid >> 5;
    float pv[WPW];
#pragma unroll
    for (int j = 0; j < WPW; ++j) {
        const int wi = wave + j * NWAVES;
        const int cell = (seg - HALO + wi * 32 + lane) & (RW - 1);  // circular wrap
        pv[j] = srow[cell];
    }
#pragma unroll
    for (int j = 0; j < WPW; ++j) {
        const int wi = wave + j * NWAVES;
        const uint32_t m = __builtin_amdgcn_ballot_w32(pv[j] != 0.0f);
        if (lane == 0) words[wi] = m;
    }
    __syncthreads();

    // Each lane loads its 9-word (288-cell) window; its 16 output cells sit at
    // bits [bitoff, bitoff+16) of window word 4 => >=128-cell halo on each side.
    uint32_t w[NWIN];
#pragma unroll
    for (int i = 0; i < NWIN; ++i) w[i] = words[(tid >> 1) + i];
    const int bitoff = (tid & 1) * 16;

    // LDS byte offset of the stage buffer (flat shared addr low 32 bits == LDS offset)
    const uint32_t lds_base = (uint32_t)(uintptr_t)(&stage[0]);
    float* outb = out + ((size_t)row * TS) * RW + seg;  // SGPR base for async stores

    if (rbyte == 110u)
        time_loop<true>(w, mk, stage, lds_base, outb, tid, bitoff);
    else
        time_loop<false>(w, mk, stage, lds_base, outb, tid, bitoff);
}

extern "C" void kernel_launch(void* const* d_in, const int* in_sizes, int n_in,
                              void* d_out, int out_size, void* d_ws, size_t ws_size,
                              hipStream_t stream) {
    (void)in_sizes; (void)n_in; (void)d_ws; (void)ws_size; (void)out_size;
    const float* state = (const float*)d_in[0];  // (16,1,32768) fp32 of {0,1}
    const float* rtab  = (const float*)d_in[1];  // 8 fp32 rule-table entries
    float* out = (float*)d_out;                  // (16,129,1,32768) fp32
    ca110_kernel<<<dim3(NBLOCKS), dim3(BLOCK), 0, stream>>>(state, rtab, out);
}